// MIL_Cluster_FC_surv_29300266893806
// MI455X (gfx1250) — compile-verified
//
#include <hip/hip_runtime.h>

// ---------------- problem constants ----------------
constexpr int kN = 8192;   // patches
constexpr int kD = 1024;   // in_dim
constexpr int kH = 512;    // hidden
constexpr int kA = 256;    // attn dim
constexpr int kC = 10;     // clusters
constexpr int kK = 4;      // classes
constexpr int kPadU  = 64;                 // rows per GEMM block (cluster pad unit)
constexpr int kNPAD  = kN + kPadU * kC;    // 8832
constexpr int kTILES = kNPAD / kPadU;      // 138

typedef float v2f __attribute__((ext_vector_type(2)));
typedef float v8f __attribute__((ext_vector_type(8)));

// ---------------- routing kernels ----------------
__global__ void k_init(int* counts, int* cursor, int* order) {
  int i = blockIdx.x * blockDim.x + threadIdx.x;
  if (i < 16) { counts[i] = 0; cursor[i] = 0; }
  if (i < kNPAD) order[i] = -1;
}

__global__ void k_count(const int* __restrict__ cid, int* counts) {
  int n = blockIdx.x * blockDim.x + threadIdx.x;
  if (n < kN) atomicAdd(&counts[cid[n]], 1);
}

__global__ void k_scan(const int* __restrict__ counts, int* padOff, int* cursor) {
  if (threadIdx.x == 0) {
    int run = 0;
    for (int c = 0; c < kC; ++c) {
      padOff[c] = run;
      cursor[c] = run;
      run += ((counts[c] + kPadU - 1) / kPadU) * kPadU;
    }
    padOff[kC] = run;
  }
}

__global__ void k_scatter(const int* __restrict__ cid, int* cursor, int* order) {
  int n = blockIdx.x * blockDim.x + threadIdx.x;
  if (n < kN) {
    int c = cid[n];
    int p = atomicAdd(&cursor[c], 1);
    order[p] = n;
  }
}

// ---------------- grouped GEMM, native f32 WMMA, 4x4 register blocking ----------
// Block: 256 threads (8 waves) computes a 64-row x 512-col tile of one cluster.
// Wave: 64 columns (4 N-tiles of 16); block rows give 4 M-tiles of 16.
// Per K-step (K+=4): 4 A-frag b64 loads + 8 B dword loads feed 16 WMMAs,
// so the streamed W operand is amortized 4x across M-tiles.
template <int KDIM, bool GATHER>
__global__ void __launch_bounds__(256)
k_gemm(const float* __restrict__ A,      // GATHER ? x [kN,KDIM] : h1 [kNPAD,KDIM]
       const float* __restrict__ W,      // [kC, KDIM, kH]
       const float* __restrict__ bias,   // [kC, kH]
       const int*   __restrict__ padOff, // [kC+1]
       const int*   __restrict__ order,  // [kNPAD]
       float* __restrict__ Out)          // [kNPAD, kH]
{
  const int row0  = blockIdx.x * kPadU;
  const int total = padOff[kC];
  if (row0 >= total) return;            // uniform exit; EXEC all-1 below

  int c = 0;
  #pragma unroll
  for (int i = 1; i < kC; ++i)
    if (row0 >= padOff[i]) c = i;

  const int lane = threadIdx.x & 31;
  const int wave = threadIdx.x >> 5;    // 0..7
  const int lm   = lane & 15;           // M (A) / N (B,C) lane index
  const int kh   = (lane >> 4) * 2;     // A/B K sub-offset: low lanes K0..1, high K2..3
  const int mof  = (lane >> 4) * 8;     // C/D: high lanes hold rows M+8
  const int colB = wave * 64 + lm;

  // per-lane gathered row pointers for the 4 M-tiles
  const float* arow[4];
  #pragma unroll
  for (int mt = 0; mt < 4; ++mt) {
    int r = GATHER ? order[row0 + mt * 16 + lm] : (row0 + mt * 16 + lm);
    if (GATHER && r < 0) r = 0;         // padding rows: garbage, masked at reduce
    arow[mt] = A + (size_t)r * KDIM + kh;
  }
  const float* __restrict__ wcol = W + (size_t)c * KDIM * kH + colB;

  v8f acc[4][4] = {};
  for (int k0 = 0; k0 < KDIM; k0 += 4) {
    v2f a[4];
    #pragma unroll
    for (int mt = 0; mt < 4; ++mt)
      a[mt] = *(const v2f*)(arow[mt] + k0);

    const float* bp = wcol + (size_t)(k0 + kh) * kH;
    v2f b[4];
    #pragma unroll
    for (int nt = 0; nt < 4; ++nt) {
      b[nt].x = bp[nt * 16];
      b[nt].y = bp[nt * 16 + kH];
    }
    __builtin_prefetch(bp + (size_t)16 * kH, 0, 3);  // stream W ahead (global_prefetch)

    #pragma unroll
    for (int mt = 0; mt < 4; ++mt)
      #pragma unroll
      for (int nt = 0; nt < 4; ++nt)
        acc[mt][nt] = __builtin_amdgcn_wmma_f32_16x16x4_f32(
            false, a[mt], false, b[nt], (short)0, acc[mt][nt], false, false);
  }

  #pragma unroll
  for (int mt = 0; mt < 4; ++mt) {
    #pragma unroll
    for (int nt = 0; nt < 4; ++nt) {
      const int col = colB + nt * 16;
      const float bv = bias[c * kH + col];
      #pragma unroll
      for (int p = 0; p < 8; ++p) {
        float v = acc[mt][nt][p] + bv;
        Out[(size_t)(row0 + mt * 16 + mof + p) * kH + col] = fmaxf(v, 0.f);
      }
    }
  }
}

// ---------------- masked per-cluster mean ----------------
__global__ void k_reduce(const float* __restrict__ h2, const int* __restrict__ counts,
                         const int* __restrict__ padOff, const int* __restrict__ order,
                         float* __restrict__ hcl) {
  int c   = blockIdx.x;
  int col = blockIdx.y * blockDim.x + threadIdx.x;
  if (col >= kH) return;
  int lo = padOff[c], hi = padOff[c + 1];
  float s = 0.f;
  for (int row = lo; row < hi; ++row)
    if (order[row] >= 0) s += h2[(size_t)row * kH + col];
  float cnt = (float)counts[c];
  hcl[c * kH + col] = s / fmaxf(cnt, 1.f);
}

// ---------------- attention head + classifier (tiny, one block) ----------------
__global__ void __launch_bounds__(512)
k_final(const float* __restrict__ hcl,
        const float* __restrict__ Wf, const float* __restrict__ bf,
        const float* __restrict__ Wa, const float* __restrict__ ba,
        const float* __restrict__ Wb, const float* __restrict__ bb,
        const float* __restrict__ Wc, const float* __restrict__ bc,
        const float* __restrict__ Wr, const float* __restrict__ br,
        const float* __restrict__ Wcls, const float* __restrict__ bcls,
        float* __restrict__ out) {
  __shared__ float s_hc[kC * kH];
  __shared__ float s_h[kC * kH];
  __shared__ float s_ag[kC * kA];
  __shared__ float s_A[kC];
  __shared__ float s_hp[kH];
  __shared__ float s_hp2[kA];
  __shared__ float s_logits[kK];
  const int t = threadIdx.x;

  for (int i = t; i < kC * kH; i += 512) s_hc[i] = hcl[i];
  __syncthreads();

  // h = relu(h_cluster @ Wf + bf)
  for (int i = t; i < kC * kH; i += 512) {
    int c = i / kH, j = i % kH;
    float s = bf[j];
    for (int k = 0; k < kH; ++k) s += s_hc[c * kH + k] * Wf[k * kH + j];
    s_h[i] = fmaxf(s, 0.f);
  }
  __syncthreads();

  // gated attention: tanh(h@Wa+ba) * sigmoid(h@Wb+bb)
  for (int i = t; i < kC * kA; i += 512) {
    int c = i / kA, j = i % kA;
    float sa = ba[j], sg = bb[j];
    for (int k = 0; k < kH; ++k) {
      float hv = s_h[c * kH + k];
      sa += hv * Wa[k * kA + j];
      sg += hv * Wb[k * kA + j];
    }
    s_ag[i] = tanhf(sa) * (1.f / (1.f + expf(-sg)));
  }
  __syncthreads();

  if (t < kC) {
    float s = bc[0];
    for (int k = 0; k < kA; ++k) s += s_ag[t * kA + k] * Wc[k];
    s_A[t] = s;
  }
  __syncthreads();

  if (t == 0) {  // softmax over 10 clusters
    float m = s_A[0];
    for (int c = 1; c < kC; ++c) m = fmaxf(m, s_A[c]);
    float sum = 0.f;
    for (int c = 0; c < kC; ++c) { s_A[c] = expf(s_A[c] - m); sum += s_A[c]; }
    for (int c = 0; c < kC; ++c) s_A[c] /= sum;
  }
  __syncthreads();

  if (t < kH) {  // h_path = A @ h
    float s = 0.f;
    for (int c = 0; c < kC; ++c) s += s_A[c] * s_h[c * kH + t];
    s_hp[t] = s;
  }
  __syncthreads();

  if (t < kA) {  // rho
    float s = br[t];
    for (int k = 0; k < kH; ++k) s += s_hp[k] * Wr[k * kA + t];
    s_hp2[t] = fmaxf(s, 0.f);
  }
  __syncthreads();

  if (t < kK) {  // classifier
    float s = bcls[t];
    for (int k = 0; k < kA; ++k) s += s_hp2[k] * Wcls[k * kK + t];
    s_logits[t] = s;
  }
  __syncthreads();

  if (t == 0) {
    float best = -1e30f; int bi = 0; float run = 1.f;
    for (int k = 0; k < kK; ++k) {
      float lg = s_logits[k];
      if (lg > best) { best = lg; bi = k; }
      float hz = 1.f / (1.f + expf(-lg));
      out[k] = hz;                  // hazards
      run *= (1.f - hz);
      out[kK + k] = run;            // survival S
    }
    out[2 * kK] = (float)bi;        // Y_hat
  }
}

// ---------------- launch ----------------
extern "C" void kernel_launch(void* const* d_in, const int* in_sizes, int n_in,
                              void* d_out, int out_size, void* d_ws, size_t ws_size,
                              hipStream_t stream) {
  const float* x    = (const float*)d_in[0];
  const int*   cid  = (const int*)  d_in[1];
  const float* W1   = (const float*)d_in[2];
  const float* b1   = (const float*)d_in[3];
  const float* W2   = (const float*)d_in[4];
  const float* b2   = (const float*)d_in[5];
  const float* Wf   = (const float*)d_in[6];
  const float* bf   = (const float*)d_in[7];
  const float* Wa   = (const float*)d_in[8];
  const float* ba   = (const float*)d_in[9];
  const float* Wb   = (const float*)d_in[10];
  const float* bb   = (const float*)d_in[11];
  const float* Wc   = (const float*)d_in[12];
  const float* bc   = (const float*)d_in[13];
  const float* Wr   = (const float*)d_in[14];
  const float* br   = (const float*)d_in[15];
  const float* Wcls = (const float*)d_in[16];
  const float* bcls = (const float*)d_in[17];
  float* out = (float*)d_out;

  // workspace carve-out: ints then floats (all 4-byte aligned)
  int* ws_i   = (int*)d_ws;
  int* counts = ws_i;             // 16
  int* padOff = ws_i + 16;        // 16 (uses [0..kC])
  int* cursor = ws_i + 32;        // 16
  int* order  = ws_i + 48;        // kNPAD
  float* h1  = (float*)(ws_i + 48 + kNPAD);
  float* h2  = h1 + (size_t)kNPAD * kH;
  float* hcl = h2 + (size_t)kNPAD * kH;

  k_init   <<<(kNPAD + 255) / 256, 256, 0, stream>>>(counts, cursor, order);
  k_count  <<<(kN + 255) / 256,    256, 0, stream>>>(cid, counts);
  k_scan   <<<1, 32, 0, stream>>>(counts, padOff, cursor);
  k_scatter<<<(kN + 255) / 256,    256, 0, stream>>>(cid, cursor, order);

  k_gemm<kD, true ><<<kTILES, 256, 0, stream>>>(x,  W1, b1, padOff, order, h1);
  k_gemm<kH, false><<<kTILES, 256, 0, stream>>>(h1, W2, b2, padOff, order, h2);

  dim3 rg(kC, kH / 256);
  k_reduce<<<rg, 256, 0, stream>>>(h2, counts, padOff, order, hcl);

  k_final<<<1, 512, 0, stream>>>(hcl, Wf, bf, Wa, ba, Wb, bb, Wc, bc,
                                 Wr, br, Wcls, bcls, out);
}